// KNNPairLayer_53712861004189
// MI455X (gfx1250) — compile-verified
//
#include <hip/hip_runtime.h>
#include <stdint.h>

#define NPTS   2048
#define NFEAT  32
#define SPLIT  16                      // candidate partitions
#define CB     (NPTS / SPLIT)          // 128 candidate rows per block (16 KB LDS)
#define QB     64                      // queries per block = threads (2 waves)
#define QBLOCKS (NPTS / QB)            // 32
// 16-byte transfers per slice: CB*NFEAT*4/16 = 1024; per thread: 1024/64 = 16
#define XFERS_PER_THREAD ((CB * NFEAT * 4 / 16) / QB)

// CDNA5 async global->LDS copy (tracked by ASYNCcnt), 16 bytes per lane.
__device__ __forceinline__ void async_copy_b128(uint32_t lds_off, const void* gaddr) {
    asm volatile("global_load_async_to_lds_b128 %0, %1, off"
                 :: "v"(lds_off), "v"(gaddr)
                 : "memory");
}

__device__ __forceinline__ void wait_async_zero() {
    asm volatile("s_wait_asynccnt 0" ::: "memory");
}

// Grid: (QBLOCKS, SPLIT). Each block: 64 queries vs one 128-row slice of feat1.
__global__ __launch_bounds__(QB)
void knn_l1_partial(const float* __restrict__ feat0,
                    const float* __restrict__ feat1,
                    float* __restrict__ pdist,   // [SPLIT][NPTS]
                    int*   __restrict__ pidx)    // [SPLIT][NPTS]
{
    __shared__ __align__(16) float tile[CB * NFEAT];   // 16 KB

    const int t = threadIdx.x;                     // 0..63
    const int q = blockIdx.x * QB + t;             // query row
    const int s = blockIdx.y;                      // candidate slice

    // Issue async loads for this block's feat1 slice into LDS.
    {
        const char* gbase = (const char*)(feat1 + (size_t)s * CB * NFEAT);
        const uint32_t lbase = (uint32_t)(uintptr_t)(&tile[0]);
        #pragma unroll
        for (int i = 0; i < XFERS_PER_THREAD; ++i) {
            int e = i * QB + t;                    // coalesced per wave
            async_copy_b128(lbase + (uint32_t)e * 16u, gbase + (size_t)e * 16u);
        }
    }

    // Load my query row into registers while the async copy is in flight.
    float4 a4[NFEAT / 4];
    {
        const float4* arow = (const float4*)(feat0 + (size_t)q * NFEAT);
        #pragma unroll
        for (int i = 0; i < NFEAT / 4; ++i) a4[i] = arow[i];
    }

    wait_async_zero();     // my slice portion landed in LDS
    __syncthreads();       // publish across both waves

    float best    = 3.402823466e38f;
    int   bestIdx = s * CB;

    for (int m = 0; m < CB; ++m) {
        const float4* row = (const float4*)(&tile[m * NFEAT]);   // LDS broadcast read
        float d = 0.0f;
        #pragma unroll
        for (int i = 0; i < NFEAT / 4; ++i) {
            float4 v = row[i];
            d += __builtin_fabsf(a4[i].x - v.x);
            d += __builtin_fabsf(a4[i].y - v.y);
            d += __builtin_fabsf(a4[i].z - v.z);
            d += __builtin_fabsf(a4[i].w - v.w);
        }
        if (d < best) { best = d; bestIdx = s * CB + m; }   // strict < => lowest index on ties
    }

    pdist[s * NPTS + q] = best;
    pidx [s * NPTS + q] = bestIdx;
}

// Single block: merge the SPLIT partials per query, global max/min reduction,
// then write matched_pos gather + normalized dist.
__global__ __launch_bounds__(1024)
void finalize_kernel(const float* __restrict__ pdist,
                     const int*   __restrict__ pidx,
                     const float* __restrict__ pos1,
                     float*       __restrict__ out)
{
    __shared__ float smax[1024];
    __shared__ float smin[1024];
    const int t = threadIdx.x;

    float dn[2];
    int   jj[2];
    #pragma unroll
    for (int k = 0; k < 2; ++k) {
        const int n = t + k * 1024;
        float best = 3.402823466e38f;
        int   bi   = 0;
        #pragma unroll
        for (int s = 0; s < SPLIT; ++s) {          // ascending s keeps lowest index on ties
            float d = pdist[s * NPTS + n];
            if (d < best) { best = d; bi = pidx[s * NPTS + n]; }
        }
        dn[k] = -best;                              // neg_d, matches reference sign
        jj[k] = bi;
    }

    smax[t] = fmaxf(dn[0], dn[1]);
    smin[t] = fminf(dn[0], dn[1]);
    __syncthreads();

    #pragma unroll
    for (int s = 512; s > 0; s >>= 1) {
        if (t < s) {
            smax[t] = fmaxf(smax[t], smax[t + s]);
            smin[t] = fminf(smin[t], smin[t + s]);
        }
        __syncthreads();
    }

    // Reference precedence: th = max(dist) - (min(dist) / 4)
    const float th = smax[0] - smin[0] * 0.25f;

    #pragma unroll
    for (int k = 0; k < 2; ++k) {
        const int n = t + k * 1024;
        const float d   = dn[k];
        const float sel = ((d < th) ? 1.0f : 0.0f) + 1e-6f;
        out[2 * NPTS + n] = d / sel;               // dist at out[4096 + n]
        const int j = jj[k];
        out[2 * n + 0] = pos1[2 * j + 0];          // matched_pos
        out[2 * n + 1] = pos1[2 * j + 1];
    }
}

extern "C" void kernel_launch(void* const* d_in, const int* in_sizes, int n_in,
                              void* d_out, int out_size, void* d_ws, size_t ws_size,
                              hipStream_t stream) {
    const float* feat0 = (const float*)d_in[0];   // (1, 2048, 32) f32
    const float* feat1 = (const float*)d_in[1];   // (1, 2048, 32) f32
    const float* pos1  = (const float*)d_in[2];   // (1, 2048, 2)  f32

    float* pdist = (float*)d_ws;                                            // SPLIT*NPTS f32 (128 KB)
    int*   pidx  = (int*)((char*)d_ws + (size_t)SPLIT * NPTS * sizeof(float)); // SPLIT*NPTS i32 (128 KB)

    dim3 grid(QBLOCKS, SPLIT);
    knn_l1_partial<<<grid, QB, 0, stream>>>(feat0, feat1, pdist, pidx);
    finalize_kernel<<<1, 1024, 0, stream>>>(pdist, pidx, pos1, (float*)d_out);
}